// GatedGCN_59167469470092
// MI455X (gfx1250) — compile-verified
//
#include <hip/hip_runtime.h>
#include <hip/hip_bf16.h>

#define LAYERS 3
#define NNODE  50000
#define NEDGE  800000
#define DIM    64
#define BN_EPS 1e-5f
#define GATE_EPS 1e-6f

typedef __attribute__((ext_vector_type(2))) float v2f;
typedef __attribute__((ext_vector_type(8))) float v8f;

// ---------------------------------------------------------------------------
// GEMM: out[rows x 64] = X[rows x 64] @ W[64 x 64] + bias[64]
// One block = 128 threads = 4 waves; each wave owns one 16-col tile of a
// 16-row tile. K=64 swept as 16 steps of V_WMMA_F32_16X16X4_F32.
// A-matrix 16x4 f32 layout: lane l holds row M=l&15; half=(l>>4): VGPR j = K
// element (4*kk + 2*half + j). B-matrix 4x16: lane l holds col N=l&15,
// VGPR j = K row (4*kk + 2*half + j). C/D: VGPR v -> M = v + 8*half.
// ---------------------------------------------------------------------------
__global__ __launch_bounds__(128) void gemm_d64_wmma(
    const float* __restrict__ X, const float* __restrict__ W,
    const float* __restrict__ bias, float* __restrict__ out)
{
    __shared__ float xs[16 * 64];
    const int tile = blockIdx.x;
    const int tid  = threadIdx.x;

    const float* Xrow = X + tile * 16 * 64;
    #pragma unroll
    for (int i = tid; i < 16 * 64; i += 128) xs[i] = Xrow[i];
    __syncthreads();

    const int lane = tid & 31;
    const int wave = tid >> 5;           // col tile 0..3
    const int col  = wave * 16 + (lane & 15);
    const int half = lane >> 4;          // 0|1
    const int mrow = lane & 15;

    v8f acc = {};
    #pragma unroll
    for (int kk = 0; kk < 16; ++kk) {
        const int k0 = kk * 4 + half * 2;
        v2f a, b;
        a.x = xs[mrow * 64 + k0];
        a.y = xs[mrow * 64 + k0 + 1];
        b.x = W[k0 * 64 + col];
        b.y = W[k0 * 64 + 64 + col];
        acc = __builtin_amdgcn_wmma_f32_16x16x4_f32(
            false, a, false, b, (short)0, acc, false, false);
    }

    const float bv = bias[col];
    #pragma unroll
    for (int v = 0; v < 8; ++v)
        out[(tile * 16 + v + half * 8) * 64 + col] = acc[v] + bv;
}

// ---------------------------------------------------------------------------
// Edge BN statistics: per-feature sum & sumsq of pf = A[s]+B[t]+C and
// pb = A[t]+B[s]+C over all E edges. stats[0:64]=sum_f [64:128]=sq_f
// [128:192]=sum_b [192:256]=sq_b
// ---------------------------------------------------------------------------
__global__ __launch_bounds__(256) void edge_stats_kernel(
    const int* __restrict__ ei, const float* __restrict__ A,
    const float* __restrict__ B, const float* __restrict__ C,
    float* __restrict__ stats)
{
    const int tid = threadIdx.x;
    const int f = tid & 63;
    const int r = tid >> 6;              // 0..3 edge sub-slot
    float sf = 0.f, qf = 0.f, sb = 0.f, qb = 0.f;

    for (int eb = blockIdx.x * 4 + r; eb < NEDGE; eb += gridDim.x * 4) {
        const int s = ei[eb];
        const int t = ei[NEDGE + eb];
        const float c  = C[eb * 64 + f];
        const float pf = A[s * 64 + f] + B[t * 64 + f] + c;
        const float pb = A[t * 64 + f] + B[s * 64 + f] + c;
        sf += pf; qf += pf * pf;
        sb += pb; qb += pb * pb;
    }

    __shared__ float red[256];
    float vals[4] = {sf, qf, sb, qb};
    #pragma unroll
    for (int q = 0; q < 4; ++q) {
        red[tid] = vals[q];
        __syncthreads();
        if (r == 0) {
            float tot = red[f] + red[64 + f] + red[128 + f] + red[192 + f];
            atomicAdd(&stats[q * 64 + f], tot);
        }
        __syncthreads();
    }
}

// ---------------------------------------------------------------------------
// Node BN statistics over pre = h_f + U + h_b.  stats[256:320]=sum,
// stats[320:384]=sumsq
// ---------------------------------------------------------------------------
__global__ __launch_bounds__(256) void node_stats_kernel(
    const float* __restrict__ hf, const float* __restrict__ U,
    const float* __restrict__ hb, float* __restrict__ stats)
{
    const int tid = threadIdx.x;
    const int f = tid & 63;
    const int r = tid >> 6;
    float s = 0.f, q = 0.f;

    for (int n = blockIdx.x * 4 + r; n < NNODE; n += gridDim.x * 4) {
        const int idx = n * 64 + f;
        const float pre = hf[idx] + U[idx] + hb[idx];
        s += pre; q += pre * pre;
    }

    __shared__ float red[256];
    float vals[2] = {s, q};
    #pragma unroll
    for (int qq = 0; qq < 2; ++qq) {
        red[tid] = vals[qq];
        __syncthreads();
        if (r == 0) {
            float tot = red[f] + red[64 + f] + red[128 + f] + red[192 + f];
            atomicAdd(&stats[256 + qq * 64 + f], tot);
        }
        __syncthreads();
    }
}

// ---------------------------------------------------------------------------
// Finalize BN groups: g=0 edge-f, g=1 edge-b (count E); g=2 node (count N).
// Writes mu at stats[384 + g*128 + f], rsqrt(var+eps) at stats[384+g*128+64+f]
// ---------------------------------------------------------------------------
__global__ __launch_bounds__(64) void finalize_kernel(float* __restrict__ stats, int g0)
{
    const int g = g0 + blockIdx.x;
    const int f = threadIdx.x;
    const float cnt = (g < 2) ? (float)NEDGE : (float)NNODE;
    const float s = stats[g * 128 + f];
    const float q = stats[g * 128 + 64 + f];
    const float mu = s / cnt;
    const float var = q / cnt - mu * mu;
    stats[384 + g * 128 + f]      = mu;
    stats[384 + g * 128 + 64 + f] = rsqrtf(var + BN_EPS);
}

// ---------------------------------------------------------------------------
// Edge apply: recompute pf/pb, BN+ReLU+e -> ne, sigmoid gate, scatter gated
// messages into h_f / h_b, write ne_f as the next e.
// ---------------------------------------------------------------------------
__global__ __launch_bounds__(256) void edge_apply_kernel(
    const int* __restrict__ ei, const float* __restrict__ A,
    const float* __restrict__ B, const float* __restrict__ C,
    const float* __restrict__ Vf, const float* __restrict__ Vb,
    const float* __restrict__ e_cur, const float* __restrict__ stats,
    const float* __restrict__ ge, const float* __restrict__ be,
    float* __restrict__ e_nxt, float* __restrict__ hf, float* __restrict__ hb)
{
    const int idx = blockIdx.x * 256 + threadIdx.x;   // < E*64
    const int eb = idx >> 6;
    const int f  = idx & 63;
    const int s = ei[eb];
    const int t = ei[NEDGE + eb];

    const float c  = C[idx];
    const float pf = A[s * 64 + f] + B[t * 64 + f] + c;
    const float pb = A[t * 64 + f] + B[s * 64 + f] + c;
    const float ev = e_cur[idx];

    const float bnf = (pf - stats[384 + f]) * stats[448 + f] * ge[f] + be[f];
    const float bnb = (pb - stats[512 + f]) * stats[576 + f] * ge[f] + be[f];
    const float nef = fmaxf(bnf, 0.f) + ev;
    const float neb = fmaxf(bnb, 0.f) + ev;

    const float sigf = 1.f / (1.f + expf(-nef));
    const float sigb = 1.f / (1.f + expf(-neb));

    const float msgf = Vf[s * 64 + f] * sigf / (sigf + GATE_EPS);
    const float msgb = Vb[s * 64 + f] * sigb / (sigb + GATE_EPS);

    e_nxt[idx] = nef;
    atomicAdd(&hf[t * 64 + f], msgf);
    atomicAdd(&hb[t * 64 + f], msgb);
}

// ---------------------------------------------------------------------------
// Node apply: h_nxt = relu(BN(h_f + U + h_b)*gamma + beta) + h_cur
// ---------------------------------------------------------------------------
__global__ __launch_bounds__(256) void node_apply_kernel(
    const float* __restrict__ hf, const float* __restrict__ U,
    const float* __restrict__ hb, const float* __restrict__ h_cur,
    const float* __restrict__ stats, const float* __restrict__ gh,
    const float* __restrict__ bh, float* __restrict__ h_nxt)
{
    const int idx = blockIdx.x * 256 + threadIdx.x;   // < N*64
    const int f = idx & 63;
    const float pre = hf[idx] + U[idx] + hb[idx];
    const float bn = (pre - stats[640 + f]) * stats[704 + f] * gh[f] + bh[f];
    h_nxt[idx] = fmaxf(bn, 0.f) + h_cur[idx];
}

// ---------------------------------------------------------------------------
extern "C" void kernel_launch(void* const* d_in, const int* in_sizes, int n_in,
                              void* d_out, int out_size, void* d_ws, size_t ws_size,
                              hipStream_t stream) {
    (void)in_sizes; (void)n_in; (void)out_size; (void)ws_size;

    const int*   ei  = (const int*)d_in[0];
    const float* h0  = (const float*)d_in[1];
    const float* e0  = (const float*)d_in[2];
    const float* WA  = (const float*)d_in[3];
    const float* bA  = (const float*)d_in[4];
    const float* WB  = (const float*)d_in[5];
    const float* bB  = (const float*)d_in[6];
    const float* WC  = (const float*)d_in[7];
    const float* bC  = (const float*)d_in[8];
    const float* WU  = (const float*)d_in[9];
    const float* bU  = (const float*)d_in[10];
    const float* WVf = (const float*)d_in[11];
    const float* bVf = (const float*)d_in[12];
    const float* WVb = (const float*)d_in[13];
    const float* bVb = (const float*)d_in[14];
    const float* gh  = (const float*)d_in[15];
    const float* bh  = (const float*)d_in[16];
    const float* ge  = (const float*)d_in[17];
    const float* be  = (const float*)d_in[18];

    const size_t ND = (size_t)NNODE * DIM;   // 3.2M
    const size_t ED = (size_t)NEDGE * DIM;   // 51.2M

    float* ws = (float*)d_ws;
    float* U_   = ws;
    float* Vf_  = ws + 1 * ND;
    float* Vb_  = ws + 2 * ND;
    float* A_   = ws + 3 * ND;
    float* B_   = ws + 4 * ND;
    float* C_   = ws + 5 * ND;
    float* hf_  = ws + 5 * ND + ED;
    float* hb_  = ws + 6 * ND + ED;
    float* hbufA = ws + 7 * ND + ED;
    float* hbufB = ws + 8 * ND + ED;
    float* ebufA = ws + 9 * ND + ED;
    float* ebufB = ws + 9 * ND + 2 * ED;
    float* stats = ws + 9 * ND + 3 * ED;     // 1024 floats

    float* h_cur = hbufA; float* h_nxt = hbufB;
    float* e_cur = ebufA; float* e_nxt = ebufB;

    hipMemcpyAsync(h_cur, h0, ND * sizeof(float), hipMemcpyDeviceToDevice, stream);
    hipMemcpyAsync(e_cur, e0, ED * sizeof(float), hipMemcpyDeviceToDevice, stream);

    const int nodeTiles = NNODE / 16;        // 3125
    const int edgeTiles = NEDGE / 16;        // 50000
    const size_t DD = (size_t)DIM * DIM;

    for (int l = 0; l < LAYERS; ++l) {
        hipMemsetAsync(stats, 0, 1024 * sizeof(float), stream);
        hipMemsetAsync(hf_, 0, ND * sizeof(float), stream);
        hipMemsetAsync(hb_, 0, ND * sizeof(float), stream);

        // 5 node GEMMs + 1 edge GEMM (f32 WMMA)
        gemm_d64_wmma<<<nodeTiles, 128, 0, stream>>>(h_cur, WU  + l * DD, bU  + l * DIM, U_);
        gemm_d64_wmma<<<nodeTiles, 128, 0, stream>>>(h_cur, WVf + l * DD, bVf + l * DIM, Vf_);
        gemm_d64_wmma<<<nodeTiles, 128, 0, stream>>>(h_cur, WVb + l * DD, bVb + l * DIM, Vb_);
        gemm_d64_wmma<<<nodeTiles, 128, 0, stream>>>(h_cur, WA  + l * DD, bA  + l * DIM, A_);
        gemm_d64_wmma<<<nodeTiles, 128, 0, stream>>>(h_cur, WB  + l * DD, bB  + l * DIM, B_);
        gemm_d64_wmma<<<edgeTiles, 128, 0, stream>>>(e_cur, WC  + l * DD, bC  + l * DIM, C_);

        // edge BN stats (forward + backward gates), then finalize groups 0,1
        edge_stats_kernel<<<2048, 256, 0, stream>>>(ei, A_, B_, C_, stats);
        finalize_kernel<<<2, 64, 0, stream>>>(stats, 0);

        // fused gate + scatter
        edge_apply_kernel<<<(NEDGE * DIM) / 256, 256, 0, stream>>>(
            ei, A_, B_, C_, Vf_, Vb_, e_cur, stats,
            ge + l * DIM, be + l * DIM, e_nxt, hf_, hb_);

        // node BN stats + finalize group 2
        node_stats_kernel<<<1024, 256, 0, stream>>>(hf_, U_, hb_, stats);
        finalize_kernel<<<1, 64, 0, stream>>>(stats, 2);

        // node update
        node_apply_kernel<<<(NNODE * DIM) / 256, 256, 0, stream>>>(
            hf_, U_, hb_, h_cur, stats, gh + l * DIM, bh + l * DIM, h_nxt);

        // ping-pong
        float* tmp;
        tmp = h_cur; h_cur = h_nxt; h_nxt = tmp;
        tmp = e_cur; e_cur = e_nxt; e_nxt = tmp;
    }

    float* out = (float*)d_out;              // [h (N*D) | e (E*D)]
    hipMemcpyAsync(out,      h_cur, ND * sizeof(float), hipMemcpyDeviceToDevice, stream);
    hipMemcpyAsync(out + ND, e_cur, ED * sizeof(float), hipMemcpyDeviceToDevice, stream);
}